// SetTransformerTorch_51058571215453
// MI455X (gfx1250) — compile-verified
//
#include <hip/hip_runtime.h>
#include <hip/hip_bf16.h>
#include <math.h>

typedef float v2f __attribute__((ext_vector_type(2)));
typedef float v8f __attribute__((ext_vector_type(8)));

#define WMMA_F32(a, b, c) \
  __builtin_amdgcn_wmma_f32_16x16x4_f32(false, (a), false, (b), (short)0, (c), false, false)

// ---------------------------------------------------------------------------
// m = F[N,D] @ Wm[H,D]^T + bm[H]
// One wave computes a 32-row x 128-col slab: 2 row sub-tiles x 8 col tiles.
// A-frag (16x4 f32): lane L holds M=L%16, K = v + 2*(L/16)  -> float2 load.
// B-frag (4x16 f32): lane L holds N=L%16, K = v + 2*(L/16)  -> float2 load of W row.
// C/D   (16x16 f32): lane L holds N=L%16, VGPR v holds M = v + 8*(L/16).
// ---------------------------------------------------------------------------
__global__ __launch_bounds__(256)
void k_m(const float* __restrict__ F, const float* __restrict__ Wm,
         const float* __restrict__ bm, float* __restrict__ M,
         int N, int D, int H) {
  const int wavesPerBlock = blockDim.x >> 5;
  int wave = blockIdx.x * wavesPerBlock + (threadIdx.x >> 5);
  int colGroups = H >> 7;                 // H/128
  int rowBlk = wave / colGroups;          // 32-row block index
  int colGroup = wave % colGroups;
  int lane = threadIdx.x & 31;
  int l16 = lane & 15;
  int khalf = (lane >> 4) << 1;           // 0 or 2

  int r0 = rowBlk * 32 + l16;
  const float* A0 = F + (size_t)r0 * D;
  const float* A1 = F + (size_t)(r0 + 16) * D;

  v8f acc[2][8];
#pragma unroll
  for (int s = 0; s < 2; ++s)
#pragma unroll
    for (int t = 0; t < 8; ++t) acc[s][t] = (v8f){};

  for (int k = 0; k < D; k += 4) {
    v2f a0 = *(const v2f*)(A0 + k + khalf);
    v2f a1 = *(const v2f*)(A1 + k + khalf);
#pragma unroll
    for (int t = 0; t < 8; ++t) {
      int col = colGroup * 128 + t * 16 + l16;
      v2f b = *(const v2f*)(Wm + (size_t)col * D + k + khalf);
      acc[0][t] = WMMA_F32(a0, b, acc[0][t]);
      acc[1][t] = WMMA_F32(a1, b, acc[1][t]);
    }
  }

  int rb = rowBlk * 32 + ((lane >> 4) << 3);
#pragma unroll
  for (int t = 0; t < 8; ++t) {
    int col = colGroup * 128 + t * 16 + l16;
    float bias = bm[col];
#pragma unroll
    for (int v = 0; v < 8; ++v) {
      M[(size_t)(rb + v) * H + col]      = acc[0][t][v] + bias;
      M[(size_t)(rb + 16 + v) * H + col] = acc[1][t][v] + bias;
    }
  }
}

// ---------------------------------------------------------------------------
// gates[G,4H] = Q[G,2H] @ Wih[4H,2H]^T + Hst[G,H] @ Whh[4H,H]^T
// Same 32x128-per-wave tiling; two K loops accumulating into the same tiles.
// ---------------------------------------------------------------------------
__global__ __launch_bounds__(256)
void k_gates(const float* __restrict__ Q, const float* __restrict__ Hst,
             const float* __restrict__ Wih, const float* __restrict__ Whh,
             float* __restrict__ gates, int G, int H) {
  const int wavesPerBlock = blockDim.x >> 5;
  int wave = blockIdx.x * wavesPerBlock + (threadIdx.x >> 5);
  int outCols = 4 * H;
  int colGroups = outCols >> 7;
  int rowBlk = wave / colGroups;
  int colGroup = wave % colGroups;
  int lane = threadIdx.x & 31;
  int l16 = lane & 15;
  int khalf = (lane >> 4) << 1;
  int K1 = 2 * H, K2 = H;

  v8f acc[2][8];
#pragma unroll
  for (int s = 0; s < 2; ++s)
#pragma unroll
    for (int t = 0; t < 8; ++t) acc[s][t] = (v8f){};

  int r0 = rowBlk * 32 + l16;
  {
    const float* A0 = Q + (size_t)r0 * K1;
    const float* A1 = Q + (size_t)(r0 + 16) * K1;
    for (int k = 0; k < K1; k += 4) {
      v2f a0 = *(const v2f*)(A0 + k + khalf);
      v2f a1 = *(const v2f*)(A1 + k + khalf);
#pragma unroll
      for (int t = 0; t < 8; ++t) {
        int col = colGroup * 128 + t * 16 + l16;
        v2f b = *(const v2f*)(Wih + (size_t)col * K1 + k + khalf);
        acc[0][t] = WMMA_F32(a0, b, acc[0][t]);
        acc[1][t] = WMMA_F32(a1, b, acc[1][t]);
      }
    }
  }
  {
    const float* A0 = Hst + (size_t)r0 * K2;
    const float* A1 = Hst + (size_t)(r0 + 16) * K2;
    for (int k = 0; k < K2; k += 4) {
      v2f a0 = *(const v2f*)(A0 + k + khalf);
      v2f a1 = *(const v2f*)(A1 + k + khalf);
#pragma unroll
      for (int t = 0; t < 8; ++t) {
        int col = colGroup * 128 + t * 16 + l16;
        v2f b = *(const v2f*)(Whh + (size_t)col * K2 + k + khalf);
        acc[0][t] = WMMA_F32(a0, b, acc[0][t]);
        acc[1][t] = WMMA_F32(a1, b, acc[1][t]);
      }
    }
  }

  int rb = rowBlk * 32 + ((lane >> 4) << 3);
#pragma unroll
  for (int t = 0; t < 8; ++t) {
    int col = colGroup * 128 + t * 16 + l16;
#pragma unroll
    for (int v = 0; v < 8; ++v) {
      gates[(size_t)(rb + v) * outCols + col]      = acc[0][t][v];
      gates[(size_t)(rb + 16 + v) * outCols + col] = acc[1][t][v];
    }
  }
}

// ---------------------------------------------------------------------------
// LSTM cell elementwise: gates (i,f,g,o) + biases -> c, h; h also into q_star[:, :H]
// ---------------------------------------------------------------------------
__global__ __launch_bounds__(256)
void k_lstm(const float* __restrict__ gates, const float* __restrict__ bih,
            const float* __restrict__ bhh, float* __restrict__ c,
            float* __restrict__ h, float* __restrict__ qstar, int G, int H) {
  int idx = blockIdx.x * blockDim.x + threadIdx.x;
  if (idx >= G * H) return;
  int g = idx / H, j = idx - g * H;
  const float* gr = gates + (size_t)g * 4 * H;
  float ig = gr[j]         + bih[j]         + bhh[j];
  float fg = gr[H + j]     + bih[H + j]     + bhh[H + j];
  float gg = gr[2 * H + j] + bih[2 * H + j] + bhh[2 * H + j];
  float og = gr[3 * H + j] + bih[3 * H + j] + bhh[3 * H + j];
  float si = 1.f / (1.f + expf(-ig));
  float sf = 1.f / (1.f + expf(-fg));
  float so = 1.f / (1.f + expf(-og));
  float cn = sf * c[idx] + si * tanhf(gg);
  c[idx] = cn;
  float hn = so * tanhf(cn);
  h[idx] = hn;
  qstar[(size_t)g * 2 * H + j] = hn;
}

// ---------------------------------------------------------------------------
// e[n] = dot(m[n,:], h[gi[n],:])  — one wave per node, shuffle reduction
// ---------------------------------------------------------------------------
__global__ __launch_bounds__(256)
void k_e(const float* __restrict__ M, const float* __restrict__ h,
         const int* __restrict__ gi, float* __restrict__ e, int N, int H) {
  int wave = blockIdx.x * (blockDim.x >> 5) + (threadIdx.x >> 5);
  if (wave >= N) return;
  int lane = threadIdx.x & 31;
  int g = gi[wave];
  const float* mr = M + (size_t)wave * H;
  const float* hr = h + (size_t)g * H;
  float s = 0.f;
  for (int j = lane; j < H; j += 32) s += mr[j] * hr[j];
#pragma unroll
  for (int off = 16; off; off >>= 1) s += __shfl_xor(s, off, 32);
  if (lane == 0) e[wave] = s;
}

__device__ __forceinline__ void graph_range(const int* __restrict__ gi, int N,
                                            int g, int& start, int& end) {
  int lo = 0, hi = N;                       // lower bound
  while (lo < hi) { int mid = (lo + hi) >> 1; if (gi[mid] < g) lo = mid + 1; else hi = mid; }
  start = lo;
  hi = N;                                   // upper bound
  while (lo < hi) { int mid = (lo + hi) >> 1; if (gi[mid] <= g) lo = mid + 1; else hi = mid; }
  end = lo;
}

// ---------------------------------------------------------------------------
// Per-graph softmax: a[n] = exp(clip(e[n]-max,-30,20)) / (sum + eps)
// gi is sorted -> one block per graph, binary-searched node range.
// ---------------------------------------------------------------------------
__global__ __launch_bounds__(256)
void k_att(const int* __restrict__ gi, const float* __restrict__ e,
           float* __restrict__ a, int N) {
  __shared__ float red[256];
  int g = blockIdx.x, start, end;
  graph_range(gi, N, g, start, end);
  if (start >= end) return;                 // empty graph: r stays 0 elsewhere
  int tid = threadIdx.x;

  float lmax = -INFINITY;
  for (int n = start + tid; n < end; n += blockDim.x) lmax = fmaxf(lmax, e[n]);
  red[tid] = lmax; __syncthreads();
  for (int s = 128; s; s >>= 1) { if (tid < s) red[tid] = fmaxf(red[tid], red[tid + s]); __syncthreads(); }
  float emax = red[0]; __syncthreads();

  float lsum = 0.f;
  for (int n = start + tid; n < end; n += blockDim.x) {
    float x = fminf(fmaxf(e[n] - emax, -30.f), 20.f);
    float ex = expf(x);
    a[n] = ex;
    lsum += ex;
  }
  red[tid] = lsum; __syncthreads();
  for (int s = 128; s; s >>= 1) { if (tid < s) red[tid] += red[tid + s]; __syncthreads(); }
  float inv = 1.f / (red[0] + 1e-7f);
  for (int n = start + tid; n < end; n += blockDim.x) a[n] *= inv;
}

// ---------------------------------------------------------------------------
// r[g,:] = sum_n a[n] * m[n,:]  -> q_star[g, H:2H]. One block per graph.
// ---------------------------------------------------------------------------
__global__ __launch_bounds__(256)
void k_r(const int* __restrict__ gi, const float* __restrict__ M,
         const float* __restrict__ a, float* __restrict__ qstar, int N, int H) {
  int g = blockIdx.x, start, end;
  graph_range(gi, N, g, start, end);
  int tid = threadIdx.x;
  float* qr = qstar + (size_t)g * 2 * H + H;
  for (int j = tid; j < H; j += blockDim.x) {
    float acc = 0.f;
    for (int n = start; n < end; ++n) acc += a[n] * M[(size_t)n * H + j];
    qr[j] = acc;
  }
}

__global__ void k_zero(float* __restrict__ p, size_t n) {
  size_t i = (size_t)blockIdx.x * blockDim.x + threadIdx.x;
  if (i < n) p[i] = 0.f;
}

extern "C" void kernel_launch(void* const* d_in, const int* in_sizes, int n_in,
                              void* d_out, int out_size, void* d_ws, size_t ws_size,
                              hipStream_t stream) {
  const float* F   = (const float*)d_in[0];   // [N, D]
  const int*   gi  = (const int*)d_in[1];     // [N] sorted
  const float* Wm  = (const float*)d_in[2];   // [H, D]
  const float* bm  = (const float*)d_in[3];   // [H]
  const float* Wih = (const float*)d_in[4];   // [4H, 2H]
  const float* Whh = (const float*)d_in[5];   // [4H, H]
  const float* bih = (const float*)d_in[6];   // [4H]
  const float* bhh = (const float*)d_in[7];   // [4H]
  float* qstar = (float*)d_out;               // [G, 2H]

  const int N = in_sizes[1];
  const int H = in_sizes[3];
  const int D = in_sizes[0] / N;
  const int G = out_size / (2 * H);           // derive G host-side (no device read)

  float* ws = (float*)d_ws;
  size_t off = 0;
  float* M     = ws + off; off += (size_t)N * H;       // 268 MB
  float* gates = ws + off; off += (size_t)G * 4 * H;   // 16 MB
  float* hbuf  = ws + off; off += (size_t)G * H;
  float* cbuf  = ws + off; off += (size_t)G * H;
  float* ebuf  = ws + off; off += (size_t)N;
  float* abuf  = ws + off; off += (size_t)N;

  // h = c = 0, q_star = 0
  {
    size_t nh = (size_t)G * H;
    k_zero<<<(unsigned)((nh + 255) / 256), 256, 0, stream>>>(hbuf, nh);
    k_zero<<<(unsigned)((nh + 255) / 256), 256, 0, stream>>>(cbuf, nh);
    size_t nq = (size_t)G * 2 * H;
    k_zero<<<(unsigned)((nq + 255) / 256), 256, 0, stream>>>(qstar, nq);
  }

  // m = F @ Wm^T + bm
  {
    int waves = (N / 32) * (H / 128);
    k_m<<<waves / 8, 256, 0, stream>>>(F, Wm, bm, M, N, D, H);
  }

  for (int it = 0; it < 3; ++it) {
    int gWaves = (G / 32) * ((4 * H) / 128);
    k_gates<<<gWaves / 8, 256, 0, stream>>>(qstar, hbuf, Wih, Whh, gates, G, H);
    k_lstm<<<(G * H + 255) / 256, 256, 0, stream>>>(gates, bih, bhh, cbuf, hbuf, qstar, G, H);
    k_e<<<(N + 7) / 8, 256, 0, stream>>>(M, hbuf, gi, ebuf, N, H);
    k_att<<<G, 256, 0, stream>>>(gi, ebuf, abuf, N);
    k_r<<<G, 256, 0, stream>>>(gi, M, abuf, qstar, N, H);
  }
}